// FusionEncoder_73924977098771
// MI455X (gfx1250) — compile-verified
//
#include <hip/hip_runtime.h>
#include <math.h>

typedef __attribute__((ext_vector_type(2))) float v2f;
typedef __attribute__((ext_vector_type(8))) float v8f;

#define E_DIM 1024
#define F_DIM 4096
#define N_DOC 64
#define L_DEP 2

// ---------------------------------------------------------------------------
// WMMA fp32 GEMM:  Y[64 x Nout] = act( X[64 x K] @ W[K x Nout, row stride ldw] + bias )
// Block = 256 threads = 8 waves; block covers 64 output columns.
// Wave w -> row tile r = w&3, col tiles c = (w>>2) and (w>>2)+2  (shared A frag).
// Uses V_WMMA_F32_16X16X4_F32 (K consumed 4 per instruction).
// ---------------------------------------------------------------------------
template <int RELU>
__global__ __launch_bounds__(256) void gemm64_wmma(
    const float* __restrict__ X, int K,
    const float* __restrict__ W, int ldw,
    const float* __restrict__ bias,
    float* __restrict__ Y, int Nout)
{
    const int tid  = threadIdx.x;
    const int wave = tid >> 5;
    const int lane = tid & 31;
    const int half = lane >> 4;      // 0: lanes 0-15, 1: lanes 16-31
    const int lr   = lane & 15;

    const int r  = wave & 3;         // row tile 0..3  (M = 64)
    const int c0 = wave >> 2;        // first col tile 0..1; second is c0+2

    const int rowA = r * 16 + lr;    // A-matrix row held by this lane
    const int colBlock = blockIdx.x * 64;
    const int col0 = colBlock + c0 * 16 + lr;
    const int col1 = colBlock + (c0 + 2) * 16 + lr;

    v8f acc0 = {};
    v8f acc1 = {};

    const float* Xrow = X + (size_t)rowA * K;

    for (int k0 = 0; k0 < K; k0 += 4) {
        const int kk = k0 + 2 * half;           // this lane's K pair base
        // A frag: VGPR0 = X[row][kk], VGPR1 = X[row][kk+1]  (contiguous -> b64 load)
        v2f a = *(const v2f*)(Xrow + kk);
        // B frag: VGPR0 = W[kk][col], VGPR1 = W[kk+1][col]
        const float* Wk = W + (size_t)kk * ldw;
        v2f b0; b0.x = Wk[col0]; b0.y = Wk[ldw + col0];
        v2f b1; b1.x = Wk[col1]; b1.y = Wk[ldw + col1];

        acc0 = __builtin_amdgcn_wmma_f32_16x16x4_f32(false, a, false, b0,
                                                     (short)0, acc0, false, false);
        acc1 = __builtin_amdgcn_wmma_f32_16x16x4_f32(false, a, false, b1,
                                                     (short)0, acc1, false, false);
    }

    const float bia0 = bias ? bias[col0] : 0.0f;
    const float bia1 = bias ? bias[col1] : 0.0f;

    // C/D layout: VGPR i holds M = i (lanes 0-15) / M = 8+i (lanes 16-31), N = lane&15
#pragma unroll
    for (int i = 0; i < 8; ++i) {
        const int row = r * 16 + i + 8 * half;
        float y0 = acc0[i] + bia0;
        float y1 = acc1[i] + bia1;
        if (RELU) { y0 = fmaxf(y0, 0.0f); y1 = fmaxf(y1, 0.0f); }
        Y[(size_t)row * Nout + col0] = y0;
        Y[(size_t)row * Nout + col1] = y1;
    }
}

// ---------------------------------------------------------------------------
// Residual + LayerNorm over E=1024, one block per row, in-place on x.
// ---------------------------------------------------------------------------
__global__ __launch_bounds__(256) void resid_ln(
    float* __restrict__ x, const float* __restrict__ res,
    const float* __restrict__ g, const float* __restrict__ b)
{
    __shared__ float red[256];
    const int n = blockIdx.x;
    const int tid = threadIdx.x;

    float s[4];
    float lsum = 0.0f;
#pragma unroll
    for (int i = 0; i < 4; ++i) {
        const int j = tid + i * 256;
        s[i] = x[(size_t)n * E_DIM + j] + res[(size_t)n * E_DIM + j];
        lsum += s[i];
    }
    red[tid] = lsum;
    __syncthreads();
    for (int off = 128; off > 0; off >>= 1) {
        if (tid < off) red[tid] += red[tid + off];
        __syncthreads();
    }
    const float mean = red[0] * (1.0f / E_DIM);
    __syncthreads();

    float lsq = 0.0f;
#pragma unroll
    for (int i = 0; i < 4; ++i) { const float d = s[i] - mean; lsq += d * d; }
    red[tid] = lsq;
    __syncthreads();
    for (int off = 128; off > 0; off >>= 1) {
        if (tid < off) red[tid] += red[tid + off];
        __syncthreads();
    }
    const float var  = red[0] * (1.0f / E_DIM);
    const float rstd = rsqrtf(var + 1e-5f);

#pragma unroll
    for (int i = 0; i < 4; ++i) {
        const int j = tid + i * 256;
        x[(size_t)n * E_DIM + j] = (s[i] - mean) * rstd * g[j] + b[j];
    }
}

// ---------------------------------------------------------------------------
// Split-K vector-matrix partials: partials[ky][col] = sum_{k in chunk} xv[k]*W[k][col]
// ---------------------------------------------------------------------------
__global__ __launch_bounds__(256) void vecmat_partial(
    const float* __restrict__ xv, const float* __restrict__ W, int ldw,
    int chunkK, int K, float* __restrict__ partials, int Ncols)
{
    const int col = blockIdx.x * 256 + threadIdx.x;
    const int k0  = blockIdx.y * chunkK;
    int k1 = k0 + chunkK; if (k1 > K) k1 = K;
    float acc = 0.0f;
    for (int k = k0; k < k1; ++k)
        acc = fmaf(xv[k], W[(size_t)k * ldw + col], acc);
    partials[(size_t)blockIdx.y * Ncols + col] = acc;
}

template <int RELU>
__global__ __launch_bounds__(256) void reduce_bias(
    const float* __restrict__ partials, int nchunks, int Ncols,
    const float* __restrict__ bias, float* __restrict__ dst)
{
    const int col = blockIdx.x * 256 + threadIdx.x;
    float s = bias[col];
    for (int c = 0; c < nchunks; ++c)
        s += partials[(size_t)c * Ncols + col];
    if (RELU) s = fmaxf(s, 0.0f);
    dst[col] = s;
}

// ---------------------------------------------------------------------------
// logits[n] = dot(A[n,:F], Wa2) + ba2   (one block per row)
// ---------------------------------------------------------------------------
__global__ __launch_bounds__(256) void logits_kernel(
    const float* __restrict__ A, const float* __restrict__ Wa2,
    const float* __restrict__ ba2, float* __restrict__ logits)
{
    __shared__ float red[256];
    const int n = blockIdx.x;
    const int tid = threadIdx.x;
    float acc = 0.0f;
    for (int k = tid; k < F_DIM; k += 256)
        acc = fmaf(A[(size_t)n * F_DIM + k], Wa2[k], acc);
    red[tid] = acc;
    __syncthreads();
    for (int off = 128; off > 0; off >>= 1) {
        if (tid < off) red[tid] += red[tid + off];
        __syncthreads();
    }
    if (tid == 0) logits[n] = red[0] + ba2[0];
}

// ---------------------------------------------------------------------------
// softmax over 64 logits; w -> w_out; fused[j] = sum_n w[n]*x[n][j]
// ---------------------------------------------------------------------------
__global__ __launch_bounds__(256) void softmax_fuse(
    const float* __restrict__ logits, const float* __restrict__ x,
    float* __restrict__ fused, float* __restrict__ w_out)
{
    __shared__ float l[N_DOC];
    __shared__ float w[N_DOC];
    const int tid = threadIdx.x;
    if (tid < N_DOC) l[tid] = logits[tid];
    __syncthreads();

    float mx = -3.0e38f;
    for (int i = 0; i < N_DOC; ++i) mx = fmaxf(mx, l[i]);
    float sum = 0.0f;
    for (int i = 0; i < N_DOC; ++i) sum += expf(l[i] - mx);
    if (tid < N_DOC) {
        const float wv = expf(l[tid] - mx) / sum;
        w[tid] = wv;
        w_out[tid] = wv;
    }
    __syncthreads();

#pragma unroll
    for (int i = 0; i < 4; ++i) {
        const int j = tid + i * 256;
        float acc = 0.0f;
        for (int n = 0; n < N_DOC; ++n)
            acc = fmaf(w[n], x[(size_t)n * E_DIM + j], acc);
        fused[j] = acc;
    }
}

// ---------------------------------------------------------------------------
extern "C" void kernel_launch(void* const* d_in, const int* in_sizes, int n_in,
                              void* d_out, int out_size, void* d_ws, size_t ws_size,
                              hipStream_t stream)
{
    const float* q      = (const float*)d_in[0];
    const float* doc    = (const float*)d_in[1];
    const float* proj_W = (const float*)d_in[2];
    const float* proj_b = (const float*)d_in[3];
    const float* Wqkv   = (const float*)d_in[4];
    const float* bqkv   = (const float*)d_in[5];
    const float* Wo     = (const float*)d_in[6];
    const float* bo     = (const float*)d_in[7];
    const float* ln1_g  = (const float*)d_in[8];
    const float* ln1_b  = (const float*)d_in[9];
    const float* ln2_g  = (const float*)d_in[10];
    const float* ln2_b  = (const float*)d_in[11];
    const float* W1     = (const float*)d_in[12];
    const float* b1     = (const float*)d_in[13];
    const float* W2     = (const float*)d_in[14];
    const float* b2     = (const float*)d_in[15];
    const float* Wa1    = (const float*)d_in[16];
    const float* ba1    = (const float*)d_in[17];
    const float* Wa2    = (const float*)d_in[18];
    const float* ba2    = (const float*)d_in[19];
    const float* Wf1    = (const float*)d_in[20];
    const float* bf1    = (const float*)d_in[21];
    const float* Wf2    = (const float*)d_in[22];
    const float* bf2    = (const float*)d_in[23];

    float* out = (float*)d_out;           // [0:4096] fused_repr, [4096:4160] w
    float* ws  = (float*)d_ws;

    float* x     = ws;                    // 64*1024
    float* vbuf  = ws + 65536;            // 64*1024
    float* tbuf  = ws + 131072;           // 64*1024 (attn / h2)
    float* hbuf  = ws + 196608;           // 64*4096 (h1 / a1)
    float* qproj = ws + 458752;           // 1024
    float* logit = ws + 459776;           // 64
    float* fused = ws + 459904;           // 1024
    float* tf1   = ws + 460928;           // 4096
    float* parts = ws + 465024;           // up to 8*4096

    // 1) qproj = q @ proj_W[0:E,:] + proj_b   (split-K vec-mat)
    vecmat_partial<<<dim3(4, 2), 256, 0, stream>>>(q, proj_W, E_DIM, 512, E_DIM,
                                                   parts, E_DIM);
    reduce_bias<0><<<4, 256, 0, stream>>>(parts, 2, E_DIM, proj_b, qproj);

    // 2) x = doc @ proj_W[E:2E,:] + qproj
    gemm64_wmma<0><<<16, 256, 0, stream>>>(doc, E_DIM,
                                           proj_W + (size_t)E_DIM * E_DIM, E_DIM,
                                           qproj, x, E_DIM);

    for (int l = 0; l < L_DEP; ++l) {
        const size_t wqkv_off = (size_t)l * E_DIM * 3 * E_DIM + 2 * E_DIM;
        // v = x @ Wqkv[l][:, 2E:3E] + bqkv[l][2E:]   (skip unused Q/K slices)
        gemm64_wmma<0><<<16, 256, 0, stream>>>(x, E_DIM,
                                               Wqkv + wqkv_off, 3 * E_DIM,
                                               bqkv + (size_t)l * 3 * E_DIM + 2 * E_DIM,
                                               vbuf, E_DIM);
        // attn = v @ Wo[l] + bo[l]
        gemm64_wmma<0><<<16, 256, 0, stream>>>(vbuf, E_DIM,
                                               Wo + (size_t)l * E_DIM * E_DIM, E_DIM,
                                               bo + (size_t)l * E_DIM,
                                               tbuf, E_DIM);
        // x = LN(x + attn)
        resid_ln<<<64, 256, 0, stream>>>(x, tbuf,
                                         ln1_g + (size_t)l * E_DIM,
                                         ln1_b + (size_t)l * E_DIM);
        // h1 = relu(x @ W1[l] + b1[l])
        gemm64_wmma<1><<<64, 256, 0, stream>>>(x, E_DIM,
                                               W1 + (size_t)l * E_DIM * F_DIM, F_DIM,
                                               b1 + (size_t)l * F_DIM,
                                               hbuf, F_DIM);
        // h2 = h1 @ W2[l] + b2[l]
        gemm64_wmma<0><<<16, 256, 0, stream>>>(hbuf, F_DIM,
                                               W2 + (size_t)l * F_DIM * E_DIM, E_DIM,
                                               b2 + (size_t)l * E_DIM,
                                               tbuf, E_DIM);
        // x = LN(x + h2)
        resid_ln<<<64, 256, 0, stream>>>(x, tbuf,
                                         ln2_g + (size_t)l * E_DIM,
                                         ln2_b + (size_t)l * E_DIM);
    }

    // a1 = relu(x @ Wa1 + ba1)
    gemm64_wmma<1><<<64, 256, 0, stream>>>(x, E_DIM, Wa1, F_DIM, ba1, hbuf, F_DIM);

    // logits, softmax, fused pooling (w written straight to out[4096:])
    logits_kernel<<<64, 256, 0, stream>>>(hbuf, Wa2, ba2, logit);
    softmax_fuse<<<1, 256, 0, stream>>>(logit, x, fused, out + 4096);

    // t = relu(fused @ Wf1 + bf1)
    vecmat_partial<<<dim3(16, 4), 256, 0, stream>>>(fused, Wf1, F_DIM, 256, E_DIM,
                                                    parts, F_DIM);
    reduce_bias<1><<<16, 256, 0, stream>>>(parts, 4, F_DIM, bf1, tf1);

    // out[0:4096] = t @ Wf2 + bf2
    vecmat_partial<<<dim3(16, 8), 256, 0, stream>>>(tf1, Wf2, F_DIM, 512, F_DIM,
                                                    parts, F_DIM);
    reduce_bias<0><<<16, 256, 0, stream>>>(parts, 8, F_DIM, bf2, out);
}